// DiffOfGaussians_52965536694752
// MI455X (gfx1250) — compile-verified
//
#include <hip/hip_runtime.h>

typedef __attribute__((ext_vector_type(2))) float v2f;
typedef __attribute__((ext_vector_type(8))) float v8f;

#define UNITS 256
#define BATCH 64
#define HH    128
#define WW    128
#define CC    16
#define KTOT  (HH * WW)       // 16384
#define GK    256             // K-chunk workgroups
#define KR    (KTOT / GK)     // 64 k-values per WG
#define KC    32              // k-values per LDS stage
#define NSUB  (KR / KC)       // 2 stages per WG
#define KSTR  34              // k-major row stride (even for b64 align; gcd(34,64)=2 -> conflict-free)
#define TWO_PI 6.28318530717958647692f

// partial[g][b][u] += S_slice(g) x F_slice(g); one WG per K-slice.
__global__ __launch_bounds__(256) void dog_main(
    const float* __restrict__ in,
    const float* __restrict__ a1, const float* __restrict__ a2,
    const float* __restrict__ s1, const float* __restrict__ s2,
    const float* __restrict__ ux, const float* __restrict__ uy,
    float* __restrict__ partial, float* __restrict__ out, int use_ws)
{
  __shared__ float Slds[BATCH][KSTR];   //  8704 B : S[b][k], A operand, k-contiguous
  __shared__ float Flds[UNITS][KSTR];   // 34816 B : F[u][k], B operand, k-contiguous

  const int tid = threadIdx.x;
  const int g   = blockIdx.x;

  // Per-unit constants; blockDim.x == UNITS so u == tid.
  const float sa  = s1[tid];
  const float sb  = sa + s2[tid];
  const float c1  = a1[tid] / (TWO_PI * sa);
  const float e1  = -0.5f / sa;            // exp(d2 * e1)  (ref divides by sigma)
  const float c2  = a2[tid] / (TWO_PI * sb);
  const float e2  = -0.5f / sb;
  const float cux = ux[tid];
  const float cuy = uy[tid];

  const int wave   = tid >> 5;
  const int lane   = tid & 31;
  const int hi     = lane >> 4;       // half-wave select for fragment k-rows
  const int ln     = lane & 15;       // M/N index within a tile
  const int mt     = wave >> 1;       // 0..3   -> batch tile
  const int ntbase = (wave & 1) * 8;  // 0 or 8 -> first of 8 unit tiles

  v8f acc[8] = {};                    // 8 x (16x16 f32) accumulators per wave

  for (int sub = 0; sub < NSUB; ++sub) {
    const int kbase = g * KR + sub * KC;
    if (sub) __syncthreads();

    // ---- Stage S into LDS: S[b][kk] = sum_c inputs[b][kbase+kk][c] ----
    // Wave-contiguous in kk => each wave streams 2 KB contiguous globals per b,
    // and writes a contiguous 128 B LDS row segment (conflict-free).
#pragma unroll
    for (int p = 0; p < (KC * BATCH) / 256; ++p) {        // 8 iters
      const int idx = tid + p * 256;
      const int b   = idx >> 5;                           // / KC
      const int kk  = idx & (KC - 1);
      const float4* src =
          (const float4*)(in + ((size_t)b * KTOT + (kbase + kk)) * CC);
      float4 v0 = src[0], v1 = src[1], v2 = src[2], v3 = src[3];
      Slds[b][kk] = (v0.x + v0.y + v0.z + v0.w) + (v1.x + v1.y + v1.z + v1.w)
                  + (v2.x + v2.y + v2.z + v2.w) + (v3.x + v3.y + v3.z + v3.w);
      if (sub + 1 < NSUB)   // pull next K-stage toward the caches
        __builtin_prefetch(in + ((size_t)b * KTOT + (kbase + KC + kk)) * CC, 0, 0);
    }

    // ---- Stage F into LDS: F[u][kk], u == tid (32 even banks per store) ----
#pragma unroll
    for (int p = 0; p < KC; ++p) {
      const int   k  = kbase + p;
      const float x  = (float)(k & (WW - 1));
      const float y  = (float)(k >> 7);
      const float dx = x - cux, dy = y - cuy;
      const float d2 = dx * dx + dy * dy;
      Flds[tid][p] = c1 * __expf(d2 * e1) - c2 * __expf(d2 * e2);
    }
    __syncthreads();

    // ---- WMMA: acc += A(16x4) x B(4x16), f32 in / f32 out ----
    // Fragment for lane(l), VGPR v: k = kk + 2*(l>>4) + v, m/n = l&15.
    // k-contiguous LDS rows => each fragment is ONE aligned ds_load_b64.
#pragma unroll
    for (int kk = 0; kk < KC; kk += 4) {
      const int k0 = kk + 2 * hi;
      const v2f av = *(const v2f*)&Slds[mt * 16 + ln][k0];
#pragma unroll
      for (int i = 0; i < 8; ++i) {
        const v2f bv = *(const v2f*)&Flds[(ntbase + i) * 16 + ln][k0];
        acc[i] = __builtin_amdgcn_wmma_f32_16x16x4_f32(
            false, av, false, bv, (short)0, acc[i], false, false);
      }
    }
  }

  // ---- Epilogue: C/D layout D[m = j + 8*hi][n = l&15] in VGPR j ----
#pragma unroll
  for (int i = 0; i < 8; ++i) {
    const int n = (ntbase + i) * 16 + ln;
#pragma unroll
    for (int j = 0; j < 8; ++j) {
      const int m = mt * 16 + j + 8 * hi;
      const float v = acc[i][j];
      if (use_ws) partial[((size_t)g * BATCH + m) * UNITS + n] = v;
      else        atomicAdd(&out[m * UNITS + n], v);
    }
  }
}

// out[b][u] = bias[u] + sum_g partial[g][b][u]
__global__ __launch_bounds__(256) void dog_reduce(
    const float* __restrict__ partial, const float* __restrict__ bias,
    float* __restrict__ out)
{
  const int i = blockIdx.x * blockDim.x + threadIdx.x;   // 0..16383
  float s = bias[i & (UNITS - 1)];
  for (int g2 = 0; g2 < GK; ++g2)
    s += partial[(size_t)g2 * (BATCH * UNITS) + i];
  out[i] = s;
}

// atomics fallback path: seed out with bias
__global__ __launch_bounds__(256) void dog_init(
    const float* __restrict__ bias, float* __restrict__ out)
{
  const int i = blockIdx.x * blockDim.x + threadIdx.x;
  out[i] = bias[i & (UNITS - 1)];
}

extern "C" void kernel_launch(void* const* d_in, const int* in_sizes, int n_in,
                              void* d_out, int out_size, void* d_ws, size_t ws_size,
                              hipStream_t stream)
{
  const float* in   = (const float*)d_in[0];
  const float* a1   = (const float*)d_in[1];
  const float* a2   = (const float*)d_in[2];
  const float* s1   = (const float*)d_in[3];
  const float* s2   = (const float*)d_in[4];
  const float* ux   = (const float*)d_in[5];
  const float* uy   = (const float*)d_in[6];
  const float* bias = (const float*)d_in[7];
  float* out     = (float*)d_out;
  float* partial = (float*)d_ws;

  const size_t need = (size_t)GK * BATCH * UNITS * sizeof(float);  // 16 MB
  const int use_ws  = (ws_size >= need) ? 1 : 0;

  if (!use_ws)
    dog_init<<<(BATCH * UNITS) / 256, 256, 0, stream>>>(bias, out);

  dog_main<<<GK, 256, 0, stream>>>(in, a1, a2, s1, s2, ux, uy,
                                   partial, out, use_ws);

  if (use_ws)
    dog_reduce<<<(BATCH * UNITS) / 256, 256, 0, stream>>>(partial, bias, out);
}